// Actor_68521908240570
// MI455X (gfx1250) — compile-verified
//
#include <hip/hip_runtime.h>
#include <hip/hip_bf16.h>

typedef __attribute__((ext_vector_type(2))) float v2f;
typedef __attribute__((ext_vector_type(8))) float v8f;

#define LUTNUM 64
#define M_OUT  512
#define FEAT   256
#define XDIM   128
#define EPOCHS 128
#define ROWS   (LUTNUM * M_OUT)   // 32768

// logp const: -log(0.5) - 0.5*log(2*pi) = 0.69314718 - 0.91893853
#define LOGC  (-0.22579135f)
// entropy: 0.5 + 0.5*log(2*pi) + log(0.5)
#define ENTC  (0.72579135f)

__device__ __forceinline__ float dist_val(float g, float m) {
    // exp( exp( -0.5*z^2 + LOGC ) ), z = (g - m)/0.5
    float z = (g - m) * 2.0f;
    return __expf(__expf(fmaf(-0.5f * z, z, LOGC)));
}

// ---------------------------------------------------------------------------
// Kernel 1: mean = tanh(x @ W + b) for both heads, via V_WMMA_F32_16X16X4_F32.
// One wave per 16x16 output tile. grid: 256 blocks x 32 threads (wave32).
// A-fragment (16x4 f32): lanes 0-15 hold K=k,k+1; lanes 16-31 hold K=k+2,k+3.
// B-fragment (4x16 f32): mirrored across N. D: vgpr v holds M=v+8*(lane>=16).
// ---------------------------------------------------------------------------
__global__ __launch_bounds__(32) void gemm_tanh_wmma(
    const float* __restrict__ x,
    const float* __restrict__ Wx, const float* __restrict__ bx,
    const float* __restrict__ Wy, const float* __restrict__ by,
    float* __restrict__ mean_base)
{
    const int tile  = blockIdx.x;       // 0..255
    const int which = tile >> 7;        // 0 = x-head, 1 = y-head
    const int t     = tile & 127;
    const int tm    = t >> 5;           // 0..3   (M tiles: 64/16)
    const int tn    = t & 31;           // 0..31  (N tiles: 512/16)

    const float* W    = which ? Wy : Wx;
    const float* bias = which ? by : bx;
    float* mean = mean_base + which * ROWS;

    const int lane = threadIdx.x;
    const int hlf  = lane >> 4;         // 0 or 1 (half-wave)
    const int l    = lane & 15;
    const int mrow = tm * 16 + l;       // A row for this lane
    const int ncol = tn * 16 + l;       // B column for this lane

    v8f acc = {};
    for (int k = 0; k < FEAT; k += 4) {
        const int ka = k + hlf * 2;
        v2f a, b;
        a.x = x[mrow * FEAT + ka];
        a.y = x[mrow * FEAT + ka + 1];
        b.x = W[ka * M_OUT + ncol];
        b.y = W[(ka + 1) * M_OUT + ncol];
        acc = __builtin_amdgcn_wmma_f32_16x16x4_f32(
            /*neg_a=*/false, a, /*neg_b=*/false, b,
            /*c_mod=*/(short)0, acc, /*reuse_a=*/false, /*reuse_b=*/false);
    }

    #pragma unroll
    for (int v = 0; v < 8; ++v) {
        const int m = tm * 16 + v + hlf * 8;
        const int n = tn * 16 + l;
        mean[m * M_OUT + n] = tanhf(acc[v] + bias[n]);  // * X_RANGE (=1)
    }
}

// ---------------------------------------------------------------------------
// Kernel 2: softmax denominators. One thread per row, 128 grid points,
// for both heads. Stores reciprocal so the gather stage multiplies.
// ---------------------------------------------------------------------------
__global__ __launch_bounds__(256) void softmax_denom(
    const float* __restrict__ mean_x, const float* __restrict__ mean_y,
    float* __restrict__ invS_x, float* __restrict__ invS_y)
{
    const int r = blockIdx.x * 256 + threadIdx.x;   // 0..ROWS-1
    const float mx = mean_x[r];
    const float my = mean_y[r];
    float sx = 0.0f, sy = 0.0f;
    #pragma unroll 8
    for (int j = 0; j < XDIM; ++j) {
        const float g = fmaf((float)j, 2.0f / 127.0f, -1.0f);  // linspace(-1,1,128)
        sx += dist_val(g, mx);
        sy += dist_val(g, my);
    }
    invS_x[r] = 1.0f / sx;
    invS_y[r] = 1.0f / sy;
}

// ---------------------------------------------------------------------------
// Kernel 3: per-(epoch,row) gather: alp[e,r] = dist(grid[action[e,r]], m_r)/S_r.
// Memory-bound: 67 MB int64 actions in, 33.5 MB f32 out, all coalesced.
// Thread 0 writes the two entropy constants.
// ---------------------------------------------------------------------------
__global__ __launch_bounds__(256) void gather_alp(
    const long long* __restrict__ action_x, const long long* __restrict__ action_y,
    const float* __restrict__ mean_x, const float* __restrict__ mean_y,
    const float* __restrict__ invS_x, const float* __restrict__ invS_y,
    float* __restrict__ out)
{
    const int t = blockIdx.x * 256 + threadIdx.x;   // 0 .. EPOCHS*ROWS-1
    const int r = t & (ROWS - 1);

    const float gx = fmaf((float)(int)action_x[t], 2.0f / 127.0f, -1.0f);
    const float gy = fmaf((float)(int)action_y[t], 2.0f / 127.0f, -1.0f);

    out[t]                  = dist_val(gx, mean_x[r]) * invS_x[r];
    out[EPOCHS * ROWS + t]  = dist_val(gy, mean_y[r]) * invS_y[r];

    if (t == 0) {
        out[2 * EPOCHS * ROWS]     = ENTC;
        out[2 * EPOCHS * ROWS + 1] = ENTC;
    }
}

extern "C" void kernel_launch(void* const* d_in, const int* in_sizes, int n_in,
                              void* d_out, int out_size, void* d_ws, size_t ws_size,
                              hipStream_t stream) {
    const float*     x  = (const float*)d_in[0];
    const float*     Wx = (const float*)d_in[1];
    const float*     bx = (const float*)d_in[2];
    const float*     Wy = (const float*)d_in[3];
    const float*     by = (const float*)d_in[4];
    const long long* ax = (const long long*)d_in[5];
    const long long* ay = (const long long*)d_in[6];
    float* out = (float*)d_out;

    // workspace: [mean_x | mean_y | invS_x | invS_y], 4 * 32768 f32 = 512 KB
    float* ws       = (float*)d_ws;
    float* mean_x   = ws;
    float* mean_y   = ws + ROWS;
    float* invS_x   = ws + 2 * ROWS;
    float* invS_y   = ws + 3 * ROWS;

    gemm_tanh_wmma<<<256, 32, 0, stream>>>(x, Wx, bx, Wy, by, ws);
    softmax_denom<<<ROWS / 256, 256, 0, stream>>>(mean_x, mean_y, invS_x, invS_y);
    gather_alp<<<(EPOCHS * ROWS) / 256, 256, 0, stream>>>(
        ax, ay, mean_x, mean_y, invS_x, invS_y, out);
}